// Canny_10737418240389
// MI455X (gfx1250) — compile-verified
//
#include <hip/hip_runtime.h>

// ---------------------------------------------------------------------------
// Canny edge detection, MI455X (gfx1250).
// Bandwidth-bound stencil (~16MB @ 23.3 TB/s => ~1us): no WMMA justified.
// CDNA5 paths used: TDM tensor_load_to_lds (3D halo tile -> LDS, TENSORcnt),
// wave32 ballot bitmasks, carry-lookahead hysteresis scan (adder identity).
// ---------------------------------------------------------------------------

typedef __attribute__((ext_vector_type(4))) unsigned int v4u;
typedef __attribute__((ext_vector_type(8))) int          v8i;
typedef __attribute__((ext_vector_type(4))) int          v4i;

#define IMG_H 1024
#define IMG_W 1024
#define TILE  32
#define GDIM  36   // gray tile  = TILE + 4 (double halo)
#define MDIM  34   // mag tile   = TILE + 2 (single halo)
#define GPLANE (GDIM * GDIM)

// ---------------------------------------------------------------------------
// Stage 1: gray -> Sobel -> mag/angle -> NMS -> thresholds -> bit rows.
// One block = 32x32 output tile, 256 threads (8 waves, wave32).
// ---------------------------------------------------------------------------
__global__ __launch_bounds__(256)
void canny_stage1(const float* __restrict__ x,       // (3, H, W) f32
                  unsigned* __restrict__ strongBits, // H * 32 words
                  unsigned* __restrict__ weakBits)   // H * 32 words
{
    __shared__ float rgb[3 * GPLANE];   // [c][y][x], y/x origin = (r0-2, c0-2)
    __shared__ float gray[GPLANE];
    __shared__ float mag[MDIM * MDIM];  // origin = (r0-1, c0-1)

    const int tid  = threadIdx.x;
    const int lane = tid & 31;
    const int wv   = tid >> 5;
    const int bx = blockIdx.x, by = blockIdx.y;
    const int r0 = by * TILE, c0 = bx * TILE;

    // Interior blocks: entire 36x36 halo is in-bounds -> TDM path.
    const bool interior = (bx >= 1) && (bx <= (IMG_W / TILE) - 2) &&
                          (by >= 1) && (by <= (IMG_H / TILE) - 2);

    if (interior) {
        if (tid < 32) {  // wave 0 issues the tensor DMA (wave-level op)
            unsigned long long ga =
                (unsigned long long)(const void*)(x + (size_t)(r0 - 2) * IMG_W + (c0 - 2));
            unsigned ldsAddr = (unsigned)(unsigned long long)(void*)&rgb[0];

            // D# group 0: count=1, lds_addr, 57-bit global_addr, type=2.
            v4u g0;
            g0[0] = 1u;
            g0[1] = ldsAddr;
            g0[2] = (unsigned)(ga & 0xFFFFFFFFull);
            g0[3] = (unsigned)((ga >> 32) & 0x1FFFFFFull) | (2u << 30);

            // D# group 1: data_size=4B; tensor 1024x1024 planes, row stride
            // 1024, plane (dim1) stride 1024*1024; tile 36x36x3.
            v8i g1;
            g1[0] = (int)(2u << 16);                                   // data_size=2 (4B)
            g1[1] = (int)((IMG_W & 0xFFFF) << 16);                     // tensor_dim0 lo
            g1[2] = (int)((IMG_W >> 16) | ((IMG_H & 0xFFFF) << 16));   // dim0 hi | dim1 lo
            g1[3] = (int)((IMG_H >> 16) | (GDIM << 16));               // dim1 hi | tile_dim0
            g1[4] = (int)(GDIM | (3 << 16));                           // tile_dim1 | tile_dim2
            g1[5] = (int)IMG_W;                                        // dim0_stride lo32
            g1[6] = 0;                                                 // dim0_stride hi | dim1_stride lo16
            g1[7] = (int)((IMG_H * IMG_W) >> 16);                      // dim1_stride [47:16]

            v4i g2 = {3, 0, 0, 0};   // tensor_dim2 = 3 (channels)
            v4i g3 = {0, 0, 0, 0};
            v8i g4 = {0, 0, 0, 0, 0, 0, 0, 0};  // extra operand on clang-23 lane

            __builtin_amdgcn_tensor_load_to_lds(g0, g1, g2, g3, g4, 0);
            __builtin_amdgcn_s_wait_tensorcnt(0);
        }
    } else {
        // Border blocks: guarded loads with zero padding (matches conv pad).
        for (int t = tid; t < 3 * GPLANE; t += 256) {
            int c   = t / GPLANE;
            int rem = t - c * GPLANE;
            int yy  = rem / GDIM, xx = rem - (rem / GDIM) * GDIM;
            int gr = r0 - 2 + yy, gc = c0 - 2 + xx;
            float v = 0.0f;
            if (gr >= 0 && gr < IMG_H && gc >= 0 && gc < IMG_W)
                v = x[(size_t)c * IMG_H * IMG_W + (size_t)gr * IMG_W + gc];
            rgb[t] = v;
        }
    }
    __syncthreads();

    // Grayscale: 0.2989 R + 0.587 G + 0.114 B
    for (int t = tid; t < GPLANE; t += 256)
        gray[t] = 0.2989f * rgb[t] + 0.587f * rgb[GPLANE + t] + 0.114f * rgb[2 * GPLANE + t];
    __syncthreads();

#define GA(y, xk) gray[(y) * GDIM + (xk)]
    // Sobel magnitude over the 34x34 halo region.
    for (int t = tid; t < MDIM * MDIM; t += 256) {
        int my = t / MDIM, mx = t - my * MDIM;
        float gxv = (GA(my, mx + 2) + 2.0f * GA(my + 1, mx + 2) + GA(my + 2, mx + 2))
                  - (GA(my, mx)     + 2.0f * GA(my + 1, mx)     + GA(my + 2, mx));
        float gyv = (GA(my, mx)     + 2.0f * GA(my, mx + 1)     + GA(my, mx + 2))
                  - (GA(my + 2, mx) + 2.0f * GA(my + 2, mx + 1) + GA(my + 2, mx + 2));
        mag[t] = sqrtf(gxv * gxv + gyv * gyv);
    }
    __syncthreads();

    // NMS + thresholds; pack per-row 32-bit masks via wave32 ballot.
    // Wave wv handles tile rows wv + 8k; lane = column within the tile.
#define MG(di, dj) mag[(rloc + 1 + (di)) * MDIM + (cloc + 1 + (dj))]
    for (int k = 0; k < 4; ++k) {
        int rloc = wv + 8 * k;
        int cloc = lane;
        int r = r0 + rloc, c = c0 + cloc;

        float cm = MG(0, 0);
        // Recompute angle at the center from the gray tile (cheap VALU).
        int my = rloc + 1, mx = cloc + 1;
        float gxv = (GA(my, mx + 2) + 2.0f * GA(my + 1, mx + 2) + GA(my + 2, mx + 2))
                  - (GA(my, mx)     + 2.0f * GA(my + 1, mx)     + GA(my + 2, mx));
        float gyv = (GA(my, mx)     + 2.0f * GA(my, mx + 1)     + GA(my, mx + 2))
                  - (GA(my + 2, mx) + 2.0f * GA(my + 2, mx + 1) + GA(my + 2, mx + 2));
        float ai = atan2f(gyv, gxv) * 57.29577951308232f;

        bool keep;
        if (ai < -22.5f || ai >= 157.5f) keep = (cm >= MG(0, -1)) && (cm >= MG(0, 1));
        else if (ai < 22.5f)             keep = (cm >= MG(-1, 0)) && (cm >= MG(1, 0));
        else if (ai < 67.5f)             keep = (cm >= MG(-1, -1)) && (cm >= MG(1, 1));
        else if (ai < 112.5f)            keep = (cm >= MG(-1, 0)) && (cm >= MG(1, 0));
        else                             keep = (cm >= MG(-1, 1)) && (cm >= MG(1, -1));

        bool validPix = (r >= 1) && (r <= IMG_H - 2) && (c >= 1) && (c <= IMG_W - 2);
        float supp = (keep && validPix) ? cm : 0.0f;
        bool sb = (supp >= 50.0f);
        bool wb = (supp >= 20.0f) && !sb;

        unsigned smask = __builtin_amdgcn_ballot_w32(sb);
        unsigned wmask = __builtin_amdgcn_ballot_w32(wb);
        if (lane == 0) {
            strongBits[r * 32 + bx] = smask;
            weakBits[r * 32 + bx]   = wmask;
        }
    }
#undef MG
#undef GA
}

// ---------------------------------------------------------------------------
// Stage 2: raster-scan hysteresis, bit-exact with the reference scan.
// Recurrence u[j] = weak[j] & (seed[j] | u[j-1])  ==  adder carry chain
// (generate = seed & weak, propagate = weak). One wave32 holds a 1024-px row
// as 32 words; intra-word via one 32-bit add, inter-word via (G,P) scan.
// ---------------------------------------------------------------------------
__global__ __launch_bounds__(32)
void canny_track(unsigned* __restrict__ S, const unsigned* __restrict__ Wk)
{
    __shared__ unsigned mBuf[32];
    __shared__ unsigned Gs[32];
    __shared__ unsigned Ps[32];
    const int w = threadIdx.x;  // word index: covers columns [32w, 32w+31]

    for (int i = 1; i <= IMG_H - 2; ++i) {
        unsigned prev = S[(i - 1) * 32 + w];  // row i-1: already updated
        unsigned curS = S[i * 32 + w];        // row i:   original strong
        unsigned nxt  = S[(i + 1) * 32 + w];  // row i+1: original strong
        unsigned weak = Wk[i * 32 + w];

        // Seed = any strong (updated above / original here & below) in the
        // 3x3 neighborhood, excluding the left-propagation term.
        unsigned M = prev | curS | nxt;       // own-column curS is harmless: weak&strong==0
        mBuf[w] = M;
        __syncthreads();
        unsigned ml = (w > 0) ? mBuf[w - 1] : 0u;
        unsigned mh = (w < 31) ? mBuf[w + 1] : 0u;
        __syncthreads();
        unsigned seed = M | (M << 1) | (ml >> 31) | (M >> 1) | (mh << 31);

        unsigned p = weak;
        unsigned g = seed & weak;

        // Per-word carry generate/propagate for the cross-word scan.
        unsigned s0 = p + g;
        unsigned Gc = (s0 < p) ? 1u : 0u;           // carry-out with cin=0
        unsigned Pc = (s0 == 0xFFFFFFFFu) ? 1u : 0u;
        Gs[w] = Gc; Ps[w] = Pc;
        __syncthreads();

        // Kogge-Stone inclusive scan over (G,P) across the 32 words.
        for (int d = 1; d < 32; d <<= 1) {
            unsigned gl = (w >= d) ? Gs[w - d] : 0u;
            unsigned pl = (w >= d) ? Ps[w - d] : 1u;
            __syncthreads();
            Gc = Gc | (Pc & gl);
            Pc = Pc & pl;
            Gs[w] = Gc; Ps[w] = Pc;
            __syncthreads();
        }
        unsigned cin = (w > 0) ? Gs[w - 1] : 0u;

        // carriesIn = p ^ g ^ (p + g + cin): bit b = carry into bit b.
        unsigned long long t = (unsigned long long)p + (unsigned long long)g +
                               (unsigned long long)cin;
        unsigned sum  = (unsigned)t;
        unsigned cout = (unsigned)(t >> 32);
        unsigned carriesIn = p ^ g ^ sum;
        unsigned u = (carriesIn >> 1) | (cout << 31);  // u[col b] = carry into b+1

        S[i * 32 + w] = curS | u;
        __syncthreads();
    }
}

// ---------------------------------------------------------------------------
// Stage 3: expand strong bitmap to float output (1,1,H,W).
// ---------------------------------------------------------------------------
__global__ __launch_bounds__(256)
void canny_expand(const unsigned* __restrict__ S, float* __restrict__ out)
{
    int idx = blockIdx.x * 256 + threadIdx.x;
    if (idx < IMG_H * IMG_W) {
        unsigned word = S[idx >> 5];
        out[idx] = ((word >> (idx & 31)) & 1u) ? 1.0f : 0.0f;
    }
}

extern "C" void kernel_launch(void* const* d_in, const int* in_sizes, int n_in,
                              void* d_out, int out_size, void* d_ws, size_t ws_size,
                              hipStream_t stream)
{
    (void)in_sizes; (void)n_in; (void)out_size; (void)ws_size;
    const float* x = (const float*)d_in[0];
    float* out = (float*)d_out;

    unsigned* strongBits = (unsigned*)d_ws;                 // 128 KB
    unsigned* weakBits   = strongBits + IMG_H * 32;         // 128 KB

    dim3 grid1(IMG_W / TILE, IMG_H / TILE);
    canny_stage1<<<grid1, 256, 0, stream>>>(x, strongBits, weakBits);
    canny_track<<<1, 32, 0, stream>>>(strongBits, weakBits);
    canny_expand<<<(IMG_H * IMG_W) / 256, 256, 0, stream>>>(strongBits, out);
}